// TopKRouter_45114336477542
// MI455X (gfx1250) — compile-verified
//
#include <hip/hip_runtime.h>
#include <hip/hip_bf16.h>
#include <math.h>

typedef __attribute__((ext_vector_type(2))) float v2f;
typedef __attribute__((ext_vector_type(8))) float v8f;

#define D_DIM 2048
#define N_EXP 64

// One wave (32 threads) per 32-token block (two 16-row M-tiles).
// Computes a 32x64 fp32 logits block with V_WMMA_F32_16X16X4_F32:
// 8 accumulators = 2 M-tiles x 4 N-tiles (16 experts each). B fragments are
// reused across both M-tiles, halving W/L2 traffic. x is streamed with
// non-temporal loads so the hot W working set stays resident in L2.
__global__ __launch_bounds__(32) void router_wmma_topk(
    const float* __restrict__ x,   // [tokens, 2048]
    const float* __restrict__ W,   // [64, 2048]
    const float* __restrict__ b,   // [64]
    float* __restrict__ gates_out, // [tokens, 2] gates
    float* __restrict__ idx_out)   // [tokens, 2] indices as float
{
    const int lane = threadIdx.x;        // 0..31
    const int ml   = lane & 15;
    const int hi   = lane >> 4;          // 0 or 1
    const size_t blk = blockIdx.x;       // 32-token block

    // A fragment sources: rows blk*32 + {0,16} + ml, K pair selected by hi.
    const float* __restrict__ pA0 = x + (blk * 32 + 0  + (size_t)ml) * D_DIM + 2 * hi;
    const float* __restrict__ pA1 = x + (blk * 32 + 16 + (size_t)ml) * D_DIM + 2 * hi;
    // B fragment sources: expert e = n*16 + ml, K pair selected by hi.
    const float* __restrict__ pB0 = W + ((size_t)(0 * 16 + ml)) * D_DIM + 2 * hi;
    const float* __restrict__ pB1 = W + ((size_t)(1 * 16 + ml)) * D_DIM + 2 * hi;
    const float* __restrict__ pB2 = W + ((size_t)(2 * 16 + ml)) * D_DIM + 2 * hi;
    const float* __restrict__ pB3 = W + ((size_t)(3 * 16 + ml)) * D_DIM + 2 * hi;

    v8f acc00 = {}, acc01 = {}, acc02 = {}, acc03 = {};   // M-tile 0
    v8f acc10 = {}, acc11 = {}, acc12 = {}, acc13 = {};   // M-tile 1

#pragma unroll 4
    for (int k0 = 0; k0 < D_DIM; k0 += 4) {
        // x: streamed once -> non-temporal (keep W resident in L2)
        v2f a0 = __builtin_nontemporal_load((const v2f*)(pA0 + k0));
        v2f a1 = __builtin_nontemporal_load((const v2f*)(pA1 + k0));
        // W: hot in L2 -> default temporal policy
        v2f b0 = *(const v2f*)(pB0 + k0);
        v2f b1 = *(const v2f*)(pB1 + k0);
        v2f b2 = *(const v2f*)(pB2 + k0);
        v2f b3 = *(const v2f*)(pB3 + k0);
        acc00 = __builtin_amdgcn_wmma_f32_16x16x4_f32(false, a0, false, b0, (short)0, acc00, false, false);
        acc01 = __builtin_amdgcn_wmma_f32_16x16x4_f32(false, a0, false, b1, (short)0, acc01, false, false);
        acc02 = __builtin_amdgcn_wmma_f32_16x16x4_f32(false, a0, false, b2, (short)0, acc02, false, false);
        acc03 = __builtin_amdgcn_wmma_f32_16x16x4_f32(false, a0, false, b3, (short)0, acc03, false, false);
        acc10 = __builtin_amdgcn_wmma_f32_16x16x4_f32(false, a1, false, b0, (short)0, acc10, false, false);
        acc11 = __builtin_amdgcn_wmma_f32_16x16x4_f32(false, a1, false, b1, (short)0, acc11, false, false);
        acc12 = __builtin_amdgcn_wmma_f32_16x16x4_f32(false, a1, false, b2, (short)0, acc12, false, false);
        acc13 = __builtin_amdgcn_wmma_f32_16x16x4_f32(false, a1, false, b3, (short)0, acc13, false, false);
    }

    // Scatter 32x64 logits block to LDS: row r = t*16 + j + 8*hi, e = n*16 + ml.
    __shared__ float lds_logits[32 * N_EXP];
#pragma unroll
    for (int j = 0; j < 8; ++j) {
        const int m0 = j + 8 * hi;          // M-tile 0 row
        const int m1 = 16 + m0;             // M-tile 1 row
        lds_logits[m0 * N_EXP + 0 * 16 + ml] = acc00[j];
        lds_logits[m0 * N_EXP + 1 * 16 + ml] = acc01[j];
        lds_logits[m0 * N_EXP + 2 * 16 + ml] = acc02[j];
        lds_logits[m0 * N_EXP + 3 * 16 + ml] = acc03[j];
        lds_logits[m1 * N_EXP + 0 * 16 + ml] = acc10[j];
        lds_logits[m1 * N_EXP + 1 * 16 + ml] = acc11[j];
        lds_logits[m1 * N_EXP + 2 * 16 + ml] = acc12[j];
        lds_logits[m1 * N_EXP + 3 * 16 + ml] = acc13[j];
    }
    __syncthreads();

    // Every lane handles one token: top-2 + softmax over 64 experts.
    {
        const float* row = &lds_logits[lane * N_EXP];
        float best = -INFINITY, sec = -INFINITY;
        int bi = 0, si = 0;
#pragma unroll
        for (int e = 0; e < N_EXP; ++e) {
            const float v = row[e] + b[e];
            if (v > best) {
                sec = best; si = bi;
                best = v;   bi = e;
            } else if (v > sec) {
                sec = v; si = e;
            }
        }
        // softmax over [best, sec]
        const float g0 = 1.0f / (1.0f + __expf(sec - best));
        const float g1 = 1.0f - g0;

        const size_t tok = blk * 32 + (size_t)lane;
        gates_out[tok * 2 + 0] = g0;
        gates_out[tok * 2 + 1] = g1;
        idx_out[tok * 2 + 0] = (float)bi;
        idx_out[tok * 2 + 1] = (float)si;
    }
}

extern "C" void kernel_launch(void* const* d_in, const int* in_sizes, int n_in,
                              void* d_out, int out_size, void* d_ws, size_t ws_size,
                              hipStream_t stream) {
    const float* x = (const float*)d_in[0];   // [4,4096,2048] f32
    const float* W = (const float*)d_in[1];   // [64,2048] f32
    const float* b = (const float*)d_in[2];   // [64] f32

    const int tokens = in_sizes[0] / D_DIM;   // 16384
    const int blocks = tokens / 32;           // 512

    float* out   = (float*)d_out;
    float* gates = out;                       // tokens*2 floats
    float* idx   = out + (size_t)tokens * 2;  // tokens*2 "int as float"

    router_wmma_topk<<<dim3(blocks), dim3(32), 0, stream>>>(x, W, b, gates, idx);
}